// GCN_27599459844750
// MI455X (gfx1250) — compile-verified
//
#include <hip/hip_runtime.h>

typedef float v2f __attribute__((ext_vector_type(2)));
typedef float v8f __attribute__((ext_vector_type(8)));

#define NNODES   20000
#define NEDGES   320000
#define CDIM     256
#define NGRAPH   50
#define NPG      400     // nodes per graph
#define JKW      768     // 3*256

// ---------------------------------------------------------------- prep kernels
__global__ void k_zero(float* __restrict__ p, int n) {
    int i = blockIdx.x * 256 + threadIdx.x;
    if (i < n) p[i] = 0.0f;
}

__global__ void k_deg(const int* __restrict__ rows, float* __restrict__ deg) {
    int e = blockIdx.x * 256 + threadIdx.x;
    if (e < NEDGES) unsafeAtomicAdd(&deg[rows[e]], 1.0f);
}

__global__ void k_dis(float* __restrict__ deg) {   // in place: deg -> dis
    int i = blockIdx.x * 256 + threadIdx.x;
    if (i < NNODES) {
        float d = deg[i];
        deg[i] = (d > 0.0f) ? rsqrtf(fmaxf(d, 1.0f)) : 0.0f;
    }
}

__global__ void k_norm(const int* __restrict__ rows, const int* __restrict__ cols,
                       const float* __restrict__ dis, float* __restrict__ nrm) {
    int e = blockIdx.x * 256 + threadIdx.x;
    if (e < NEDGES) nrm[e] = -(dis[rows[e]] * dis[cols[e]]);
}

// ---------------------------------------------------------------- dual GEMM
// O0 = A @ W0, O1 = A @ W1   (A: [20000 x 256] row-major, W: [256 x 256] row-major)
// Block = 256 threads = 8 waves. Block covers 32 rows.
// Wave w: rows row0 = blk*32 + (w>>2)*16 ; cols col0 = (w&3)*64 (4 col tiles).
// Uses V_WMMA_F32_16X16X4_F32 (native fp32 matrix op, matches fp32 reference).
__global__ void __launch_bounds__(256)
k_dual_gemm(const float* __restrict__ A,
            const float* __restrict__ W0, const float* __restrict__ W1,
            float* __restrict__ O0, float* __restrict__ O1) {
    const int lane = threadIdx.x & 31;
    const int wave = threadIdx.x >> 5;
    const int row0 = blockIdx.x * 32 + (wave >> 2) * 16;
    const int col0 = (wave & 3) * 64;

    const int m  = lane & 15;          // row within A tile / col within B tile
    const int kh = (lane >> 4) * 2;    // K sub-offset: lanes 0-15 -> K0/K1, 16-31 -> K2/K3

    const v8f zero = {0.f,0.f,0.f,0.f,0.f,0.f,0.f,0.f};
    v8f acc0[4] = {zero, zero, zero, zero};
    v8f acc1[4] = {zero, zero, zero, zero};

    const float* arow = A + (size_t)(row0 + m) * CDIM;

    for (int k0 = 0; k0 < CDIM; k0 += 4) {
        // A fragment (16x4 fp32): VGPR0=K{0|2}, VGPR1=K{1|3} per documented layout
        const v2f a = *(const v2f*)(arow + k0 + kh);
        const int kr0 = (k0 + kh) * CDIM;
        const int kr1 = (k0 + kh + 1) * CDIM;
#pragma unroll
        for (int t = 0; t < 4; ++t) {
            const int n = col0 + t * 16 + m;
            v2f b0, b1;
            b0.x = W0[kr0 + n]; b0.y = W0[kr1 + n];
            b1.x = W1[kr0 + n]; b1.y = W1[kr1 + n];
            acc0[t] = __builtin_amdgcn_wmma_f32_16x16x4_f32(
                false, a, false, b0, (short)0, acc0[t], false, false);
            acc1[t] = __builtin_amdgcn_wmma_f32_16x16x4_f32(
                false, a, false, b1, (short)0, acc1[t], false, false);
        }
    }

    // C/D layout: VGPR v, lane L -> M = v + 8*(L>>4), N = L&15
    const int rbase = row0 + (lane >> 4) * 8;
    const int nc    = lane & 15;
#pragma unroll
    for (int t = 0; t < 4; ++t) {
#pragma unroll
        for (int v = 0; v < 8; ++v) {
            const size_t idx = (size_t)(rbase + v) * CDIM + col0 + t * 16 + nc;
            O0[idx] = acc0[t][v];
            O1[idx] = acc1[t][v];
        }
    }
}

// ---------------------------------------------------------------- SpMM scatter
// O[row[e], :] += nrm[e] * G1[col[e], :]   (thread = one edge x one float4 chunk)
__global__ void k_spmm(const int* __restrict__ rows, const int* __restrict__ cols,
                       const float* __restrict__ nrm, const float* __restrict__ G1,
                       float* __restrict__ O) {
    const size_t tid = (size_t)blockIdx.x * 256 + threadIdx.x;  // E * 64 threads
    const int e  = (int)(tid >> 6);
    const int ch = (int)(tid & 63) * 4;
    if (e >= NEDGES) return;
    const int   r = rows[e];
    const int   c = cols[e];
    const float w = nrm[e];
    const float4 g = *(const float4*)(G1 + (size_t)c * CDIM + ch);
    float* o = O + (size_t)r * CDIM + ch;
    unsafeAtomicAdd(o + 0, w * g.x);
    unsafeAtomicAdd(o + 1, w * g.y);
    unsafeAtomicAdd(o + 2, w * g.z);
    unsafeAtomicAdd(o + 3, w * g.w);
}

// ---------------------------------------------------------------- BN/ReLU epilogue
// mode 0 (layer 0): relu(bn(v));   mode 1 (layers 1,2): bn(relu(v))
__global__ void k_bn_relu(float* __restrict__ H, const float* __restrict__ gamma,
                          const float* __restrict__ beta, int mode) {
    const size_t i = (size_t)blockIdx.x * 256 + threadIdx.x;   // N*C threads
    const int c = (int)(i & (CDIM - 1));
    const float gs = gamma[c] * (1.0f / sqrtf(1.0f + 1e-5f));
    const float bb = beta[c];
    float v = H[i];
    v = (mode == 0) ? fmaxf(v * gs + bb, 0.0f) : (fmaxf(v, 0.0f) * gs + bb);
    H[i] = v;
}

// ---------------------------------------------------------------- mean pool (deterministic)
__global__ void k_pool(const float* __restrict__ H, float* __restrict__ zagg, int layer) {
    const int tid = blockIdx.x * 256 + threadIdx.x;  // G*C = 12800
    const int g = tid >> 8;
    const int c = tid & 255;
    const float* p = H + (size_t)g * NPG * CDIM + c;
    float s = 0.0f;
    for (int j = 0; j < NPG; ++j) s += p[(size_t)j * CDIM];
    zagg[(size_t)g * JKW + layer * CDIM + c] = s * (1.0f / (float)NPG);
}

// ---------------------------------------------------------------- classifier head
__global__ void k_classifier(const float* __restrict__ zagg,
                             const float* __restrict__ W1, const float* __restrict__ b1,
                             const float* __restrict__ cg, const float* __restrict__ cbe,
                             const float* __restrict__ W2, const float* __restrict__ b2,
                             float* __restrict__ logit) {
    __shared__ float zs[256];
    const int co = threadIdx.x;
    const float gs = cg[co] * (1.0f / sqrtf(1.0f + 1e-5f));
    const float bb = cbe[co];
    const float* wr = W1 + (size_t)co * JKW;
    for (int g = 0; g < NGRAPH; ++g) {
        const float* zr = zagg + (size_t)g * JKW;
        float s = b1[co];
        for (int k = 0; k < JKW; ++k) s += zr[k] * wr[k];
        zs[co] = fmaxf(s, 0.0f) * gs + bb;
        __syncthreads();
        if (co < 2) {
            const float* w2r = W2 + co * 256;
            float acc = b2[co];
            for (int k = 0; k < 256; ++k) acc += zs[k] * w2r[k];
            logit[g * 2 + co] = acc;
        }
        __syncthreads();
    }
}

// ---------------------------------------------------------------- launcher
extern "C" void kernel_launch(void* const* d_in, const int* in_sizes, int n_in,
                              void* d_out, int out_size, void* d_ws, size_t ws_size,
                              hipStream_t stream) {
    const float* x      = (const float*)d_in[0];
    const int*   ei     = (const int*)d_in[1];
    const float* cheb_w = (const float*)d_in[3];
    const float* bng    = (const float*)d_in[4];
    const float* bnb    = (const float*)d_in[5];
    const float* w1     = (const float*)d_in[6];
    const float* b1     = (const float*)d_in[7];
    const float* cg     = (const float*)d_in[8];
    const float* cbe    = (const float*)d_in[9];
    const float* w2     = (const float*)d_in[10];
    const float* b2     = (const float*)d_in[11];

    float* zagg  = (float*)d_out;                       // [50 x 768]
    float* logit = zagg + (size_t)NGRAPH * JKW;         // [50 x 2]

    float* ws   = (float*)d_ws;
    float* deg  = ws;                                   // N (deg -> dis in place)
    float* nrm  = ws + NNODES;                          // E
    float* buf0 = nrm + NEDGES;                         // N*C each
    float* buf1 = buf0 + (size_t)NNODES * CDIM;
    float* buf2 = buf1 + (size_t)NNODES * CDIM;

    const int* rows = ei;
    const int* cols = ei + NEDGES;

    k_zero<<<(NNODES + 255) / 256, 256, 0, stream>>>(deg, NNODES);
    k_deg <<<NEDGES / 256, 256, 0, stream>>>(rows, deg);
    k_dis <<<(NNODES + 255) / 256, 256, 0, stream>>>(deg);
    k_norm<<<NEDGES / 256, 256, 0, stream>>>(rows, cols, deg, nrm);

    // buffer rotation keeps GEMM in/out/g1 disjoint every layer
    const float* hin   = x;
    float* houts[3] = {buf0, buf2, buf0};
    for (int L = 0; L < 3; ++L) {
        float* hout = houts[L];
        float* g1   = buf1;
        const float* W0 = cheb_w + (size_t)(L * 2 + 0) * CDIM * CDIM;
        const float* W1c = cheb_w + (size_t)(L * 2 + 1) * CDIM * CDIM;
        // cheb(h) = h@W0 + prop(h@W1): GEMM writes hout fully (no zero pass), SpMM accumulates
        k_dual_gemm<<<NNODES / 32, 256, 0, stream>>>(hin, W0, W1c, hout, g1);
        k_spmm<<<(NEDGES * 64) / 256, 256, 0, stream>>>(rows, cols, nrm, g1, hout);
        k_bn_relu<<<(int)(((size_t)NNODES * CDIM) / 256), 256, 0, stream>>>(
            hout, bng + L * CDIM, bnb + L * CDIM, (L == 0) ? 0 : 1);
        k_pool<<<(NGRAPH * CDIM) / 256, 256, 0, stream>>>(hout, zagg, L);
        hin = hout;
    }

    k_classifier<<<1, 256, 0, stream>>>(zagg, w1, b1, cg, cbe, w2, b2, logit);
}